// GPRFWIModel_36636071034897
// MI455X (gfx1250) — compile-verified
//
#include <hip/hip_runtime.h>

// ---------------------------------------------------------------------------
// 2D TMz FDTD (2 shots, 256x256, 512 steps) as a persistent-kernel pipeline.
//
// MI455X reasoning: ~2 GFLOP total and ~1.5MB of state -> neither compute- nor
// HBM-bound; the cost is the 512-step sequential chain. So: keep each slab's
// Ez/Hx/Hy resident in LDS (101KB of the WGP's 320KB), exchange ONE two-row
// Ez ghost zone per step (redundant Hy row removes the mid-step barrier),
// fetch halos with the CDNA5 async engines (TDM tensor_load_to_lds for one
// row, per-lane global_load_async_to_lds_b32 for the other) and overlap them
// with the halo-independent Hx update. 512 device barriers total across 16
// L2-resident blocks. The 1/DX curl factor is folded into the stored Cb so
// the hot loop has zero FP divisions. WMMA is deliberately unused: no
// contraction dimension exists, and matrix ops cannot shorten the sequential
// critical path.
// ---------------------------------------------------------------------------

namespace {
constexpr int kNX = 256, kNY = 256, kIter = 512, kNRX = 64, kPML = 10;
constexpr int kSlabs  = 8;
constexpr int kRows   = kNX / kSlabs;          // 32 rows per slab
constexpr int kShots  = 2;
constexpr int kBlocks = kShots * kSlabs;       // 16 persistent blocks
constexpr int kTPB    = 256;                   // one lane per column (8 waves)

// float32 physics constants matching the reference's np.float32 casts
constexpr double dDT  = 0.95 * 0.002 / (299792458.0 * 1.4142135623730951);
constexpr float  fDT  = (float)dDT;
constexpr double dMU0 = 4e-7 * 3.141592653589793;
constexpr float  fCH  = (float)((double)fDT / (dMU0 * 0.002));   // chx == chy
constexpr float  fEPS0   = 8.8541878128e-12f;
constexpr float  fDX     = 0.002f;
constexpr float  fSigMax = (float)(0.8 * 4.0 / (376.73031366686992 * 0.002));
constexpr float  fSrcAmp = (float)(1.0 / (0.002 * 0.002));

// LDS: Ez (kRows+2 rows incl. 2 halo rows), Hy (kRows+1 rows incl. 1 redundant
// row), Hx (kRows rows). (3*kRows+3)*256 floats = 101,376 B.
constexpr int kLdsFloats = (3 * kRows + 3) * kNY;
constexpr int kLdsBytes  = kLdsFloats * (int)sizeof(float);
} // namespace

// Per-lane CDNA5 async copy global->LDS (ASYNCcnt, no VGPR round trip).
// Low 32 bits of a generic LDS pointer are the wave-relative LDS byte offset.
__device__ __forceinline__ void async_load_f32_to_lds(float* lds_dst, const float* gsrc) {
  unsigned lds_off = (unsigned)(unsigned long long)lds_dst;
  asm volatile("global_load_async_to_lds_b32 %0, %1, off"
               :: "v"(lds_off), "v"(gsrc)
               : "memory");
}

__device__ __forceinline__ void wait_asynccnt0() {
#if __has_builtin(__builtin_amdgcn_s_wait_asynccnt)
  __builtin_amdgcn_s_wait_asynccnt(0);
#else
  asm volatile("s_wait_asynccnt 0" ::: "memory");
#endif
}

__device__ __forceinline__ void wait_tensorcnt0() {
#if __has_builtin(__builtin_amdgcn_s_wait_tensorcnt)
  __builtin_amdgcn_s_wait_tensorcnt(0);
#else
  asm volatile("s_wait_tensorcnt 0" ::: "memory");
#endif
}

// Tensor Data Mover: one-instruction DMA of a 1 x nelems f32 row into LDS,
// descriptor built in SGPRs (all operands are block-uniform). TENSORcnt.
#if __has_builtin(__builtin_amdgcn_tensor_load_to_lds)
#define HAVE_TDM 1
typedef unsigned v4u __attribute__((ext_vector_type(4)));
typedef int      v8i __attribute__((ext_vector_type(8)));
typedef int      v4i __attribute__((ext_vector_type(4)));

__device__ __forceinline__ void tdm_row_load_f32(float* lds_dst, const float* gsrc, int nelems) {
  const unsigned lds_off = (unsigned)(unsigned long long)lds_dst;
  const unsigned long long ga = (unsigned long long)gsrc;
  v4u g0;
  g0.x = 1u;                                                   // count=1, user D#
  g0.y = lds_off;                                              // lds_addr
  g0.z = (unsigned)ga;                                         // global_addr[31:0]
  g0.w = (unsigned)((ga >> 32) & 0x01FFFFFFull) | (2u << 30);  // addr[56:32] | type=2
  v8i g1;
  g1[0] = (int)(2u << 16);                                     // data_size = 4B
  g1[1] = (int)(((unsigned)nelems & 0xFFFFu) << 16);           // tensor_dim0 (bits 63:48)
  g1[2] = (int)(1u << 16);                                     // tensor_dim1 = 1 (bits 95:80)
  g1[3] = (int)(((unsigned)nelems & 0xFFFFu) << 16);           // tile_dim0 (bits 127:112)
  g1[4] = 1;                                                   // tile_dim1 = 1
  g1[5] = nelems;                                              // tensor_dim0_stride
  g1[6] = 0;
  g1[7] = 0;
  v4i z4 = {0, 0, 0, 0};
#if __clang_major__ >= 23
  v8i z8 = {0, 0, 0, 0, 0, 0, 0, 0};
  __builtin_amdgcn_tensor_load_to_lds(g0, g1, z4, z4, z8, 0);
#else
  __builtin_amdgcn_tensor_load_to_lds(g0, g1, z4, z4, 0);
#endif
}
#endif

// Device-scope barrier across the 16 persistent blocks (monotonic phase
// counter in L2; counter reset each launch by the coefficient kernel).
__device__ __forceinline__ void grid_barrier(unsigned* bar, unsigned phase) {
  __syncthreads();
  if (threadIdx.x == 0) {
    __hip_atomic_fetch_add(bar, 1u, __ATOMIC_RELEASE, __HIP_MEMORY_SCOPE_AGENT);
    const unsigned target = phase * (unsigned)kBlocks;
    while (__hip_atomic_load(bar, __ATOMIC_ACQUIRE, __HIP_MEMORY_SCOPE_AGENT) < target) {
      __builtin_amdgcn_s_sleep(2);
    }
  }
  __syncthreads();
}

// Per-cell E-update coefficients interleaved as float2 {Ca, Cb/DX}: one b64
// load per cell in the hot loop and no FP division in the time loop. Also
// resets the barrier counter for this launch.
__global__ void fdtd_coeff_kernel(const float* __restrict__ er, const float* __restrict__ se,
                                  float2* __restrict__ Cab, unsigned* __restrict__ bar) {
  const int i = blockIdx.x, j = threadIdx.x;
  const float fi = (float)i, fj = (float)j;
  const float rx = fmaxf(fmaxf((float)kPML - fi, fi - (float)(kNX - 1 - kPML)), 0.0f) * (1.0f / kPML);
  const float ry = fmaxf(fmaxf((float)kPML - fj, fj - (float)(kNY - 1 - kPML)), 0.0f) * (1.0f / kPML);
  const float sig_b = fSigMax * (rx * rx * rx + ry * ry * ry);
  const int g = i * kNY + j;
  const float se_tot = se[g] + sig_b;
  const float eps    = er[g] * fEPS0;
  const float h      = se_tot * fDT / (2.0f * eps);
  const float denom  = 1.0f + h;
  const float cb     = (fDT / eps) / denom;
  float2 c;
  c.x = (1.0f - h) / denom;        // Ca
  c.y = cb / fDX;                  // Cb/DX (curl pre-scaled; DX == DY)
  Cab[g] = c;
  if (g == 0) *bar = 0u;
}

// Persistent FDTD kernel. blockIdx.x -> (shot, slab). LDS layout per slab:
//   sEz: local 0 = Ez[r0-1] (halo), 1..kRows = owned, kRows+1 = Ez[r0+kRows] (halo)
//   sHy: local h = Hy[r0-1+h], h = 0..kRows (row 0 redundant; pad rows stay 0)
//   sHx: local i = Hx[r0+i],   i = 0..kRows-1 (col 255 unused pad)
__global__ void __launch_bounds__(kTPB, 1)
fdtd_persistent_kernel(const float2* __restrict__ Cab,
                       const float* __restrict__ wave, const int* __restrict__ src_pos,
                       const int* __restrict__ rx_pos, float* __restrict__ out,
                       float* __restrict__ halo_down, float* __restrict__ halo_up,
                       unsigned* __restrict__ bar) {
  extern __shared__ float smem[];
  float* sEz = smem;                          // (kRows+2) x 256
  float* sHy = sEz + (kRows + 2) * kNY;       // (kRows+1) x 256
  float* sHx = sHy + (kRows + 1) * kNY;       // kRows x 256

  const int blk  = blockIdx.x;
  const int shot = blk / kSlabs;
  const int slab = blk % kSlabs;
  const int r0   = slab * kRows;
  const int tid  = threadIdx.x;

  for (int idx = tid; idx < kLdsFloats; idx += kTPB) smem[idx] = 0.0f;

  const int  sx = src_pos[2 * shot + 0];
  const int  sy = src_pos[2 * shot + 1];
  const bool owns_src = (sx >= r0) && (sx < r0 + kRows);           // block-uniform
  const float cb_src  = owns_src ? Cab[sx * kNY + sy].y * fDX : 0.0f;  // raw Cb

  int rxi = 0, ryi = 0; bool owns_rx = false;
  if (tid < kNRX) {
    rxi = rx_pos[2 * tid + 0];
    ryi = rx_pos[2 * tid + 1];
    owns_rx = (rxi >= r0) && (rxi < r0 + kRows);
  }

  const bool hasLower = (slab > 0);              // block-uniform
  const bool hasUpper = (slab < kSlabs - 1);     // block-uniform
  const int  hslot    = shot * kSlabs + slab;

  __syncthreads();

  for (int t = 0; t < kIter; ++t) {
    // ---- Hy update (needs both Ez halo rows; all in LDS) ----------------
    // Edge rows are block-uniform conditionals so the middle loop unrolls.
    if (hasLower) {                              // redundant row Hy[r0-1]
      sHy[tid] += fCH * (sEz[kNY + tid] - sEz[tid]);
    }
#pragma unroll 4
    for (int h = 1; h < kRows; ++h) {
      const float e0 = sEz[h * kNY + tid];
      const float e1 = sEz[(h + 1) * kNY + tid];
      sHy[h * kNY + tid] += fCH * (e1 - e0);
    }
    if (hasUpper) {                              // top owned row Hy[r0+kRows-1]
      sHy[kRows * kNY + tid] += fCH * (sEz[(kRows + 1) * kNY + tid] - sEz[kRows * kNY + tid]);
    }
    __syncthreads();

    // ---- Ez update (fields in LDS, {Ca,Cb/DX} streamed from L2 as b64) --
#pragma unroll 4
    for (int i = 0; i < kRows; ++i) {
      const int g = (r0 + i) * kNY + tid;
      const float hy1 = sHy[(i + 1) * kNY + tid];   // Hy[row]
      const float hy0 = sHy[i * kNY + tid];         // Hy[row-1] (0 at edges)
      const float hxa = (tid < kNY - 1) ? sHx[i * kNY + tid] : 0.0f;
      const float hxb = (tid > 0) ? sHx[i * kNY + tid - 1] : 0.0f;
      const float curl = (hy1 - hy0) - (hxa - hxb);        // (1/DX) folded into c.y
      const float2 c = Cab[g];
      sEz[(i + 1) * kNY + tid] = c.x * sEz[(i + 1) * kNY + tid] + c.y * curl;
      __builtin_prefetch(&Cab[g + kNY], 0, 0);
    }

    // Source injection (block-uniform guard -> only the owning block pays
    // the extra workgroup barrier).
    if (owns_src) {
      __syncthreads();
      if (tid == 0) {
        sEz[(sx - r0 + 1) * kNY + sy] -= cb_src * wave[t] * fSrcAmp;
      }
    }
    __syncthreads();

    // Receivers: owning slab scatters Ez samples straight from LDS.
    if (owns_rx) {
      out[(shot * kIter + t) * kNRX + tid] = sEz[(rxi - r0 + 1) * kNY + ryi];
    }

    // Publish both boundary rows, one device barrier, then async-pull both
    // halo rows while the halo-independent Hx update runs underneath.
    halo_down[hslot * kNY + tid] = sEz[1 * kNY + tid];          // Ez[r0]
    halo_up  [hslot * kNY + tid] = sEz[kRows * kNY + tid];      // Ez[r0+kRows-1]
    __threadfence();
    grid_barrier(bar, (unsigned)(t + 1));

    if (hasLower) {            // bottom halo via per-lane async-to-LDS
      async_load_f32_to_lds(&sEz[tid], &halo_up[(hslot - 1) * kNY + tid]);
    }
    if (hasUpper) {            // top halo via the Tensor Data Mover (wave 0)
#ifdef HAVE_TDM
      if (tid < 32) {
        tdm_row_load_f32(&sEz[(kRows + 1) * kNY], &halo_down[(hslot + 1) * kNY], kNY);
      }
#else
      async_load_f32_to_lds(&sEz[(kRows + 1) * kNY + tid],
                            &halo_down[(hslot + 1) * kNY + tid]);
#endif
    }

    // ---- Hx update for the next step (owned Ez rows only; overlaps the
    // in-flight halo transfers) ------------------------------------------
    if (tid < kNY - 1) {
#pragma unroll 4
      for (int i = 0; i < kRows; ++i) {
        const float e0 = sEz[(i + 1) * kNY + tid];
        const float e1 = sEz[(i + 1) * kNY + tid + 1];
        sHx[i * kNY + tid] -= fCH * (e1 - e0);
      }
    }

    wait_asynccnt0();
    wait_tensorcnt0();
    __syncthreads();
  }
}

extern "C" void kernel_launch(void* const* d_in, const int* in_sizes, int n_in,
                              void* d_out, int out_size, void* d_ws, size_t ws_size,
                              hipStream_t stream) {
  (void)in_sizes; (void)n_in; (void)out_size; (void)ws_size;
  const float* er   = (const float*)d_in[0];
  const float* se   = (const float*)d_in[1];
  const float* wave = (const float*)d_in[2];
  const int*   srcp = (const int*)d_in[3];
  const int*   rxp  = (const int*)d_in[4];
  float* out = (float*)d_out;

  char* ws = (char*)d_ws;
  float2*   Cab = (float2*)(ws + 0);                           // 512 KB
  float*    hdn = (float*)(ws + 512 * 1024);                   // kBlocks*256 floats
  float*    hup = (float*)(ws + 512 * 1024 + 16 * 1024);
  unsigned* bar = (unsigned*)(ws + 512 * 1024 + 32 * 1024);

  // >64KB dynamic LDS (gfx1250 WGP supports up to 320KB per workgroup)
  (void)hipFuncSetAttribute((const void*)fdtd_persistent_kernel,
                            hipFuncAttributeMaxDynamicSharedMemorySize, kLdsBytes);

  fdtd_coeff_kernel<<<dim3(kNX), dim3(kNY), 0, stream>>>(er, se, Cab, bar);
  fdtd_persistent_kernel<<<dim3(kBlocks), dim3(kTPB), kLdsBytes, stream>>>(
      Cab, wave, srcp, rxp, out, hdn, hup, bar);
}